// SparseAutoencoder_84232898609652
// MI455X (gfx1250) — compile-verified
//
#include <hip/hip_runtime.h>
#include <math.h>

#define BDIM 2048
#define DDIM 1024
#define HDIM 32768
#define KSEL 256
#define LN_EPS 1e-5f

typedef __attribute__((ext_vector_type(2))) float v2f;
typedef __attribute__((ext_vector_type(8))) float v8f;
typedef __attribute__((ext_vector_type(4))) int v4i;

#define AS1 __attribute__((address_space(1)))
#define AS3 __attribute__((address_space(3)))

#if defined(__has_builtin)
#if __has_builtin(__builtin_amdgcn_global_load_async_to_lds_b128)
#define USE_ASYNC_LDS 1
#endif
#endif

#if defined(__has_builtin)
#if __has_builtin(__builtin_amdgcn_s_wait_asynccnt)
#define WAIT_ASYNCCNT(n) __builtin_amdgcn_s_wait_asynccnt(n)
#else
#define WAIT_ASYNCCNT(n) asm volatile("s_wait_asynccnt %0" ::"n"(n) : "memory")
#endif
#else
#define WAIT_ASYNCCNT(n)
#endif

// monotonic float->uint map: larger float => larger unsigned
__device__ __forceinline__ unsigned mapf(float f) {
  unsigned b = __float_as_uint(f);
  return (b & 0x80000000u) ? ~b : (b | 0x80000000u);
}

// ------------------------------------------------------------------
// 1) LayerNorm-style preprocess: mu, std (ddof=1), xn = (x-mu)/(std+eps)
// ------------------------------------------------------------------
__global__ __launch_bounds__(256) void ln_kernel(const float* __restrict__ x,
                                                 float* __restrict__ xn,
                                                 float* __restrict__ mu,
                                                 float* __restrict__ sd) {
  const int b = blockIdx.x;
  const int tid = threadIdx.x;
  const float4 v = *(const float4*)(x + (size_t)b * DDIM + tid * 4);
  float s = v.x + v.y + v.z + v.w;
  float q = v.x * v.x + v.y * v.y + v.z * v.z + v.w * v.w;
  __shared__ float r1[256], r2[256];
  __shared__ float sMu, sInv;
  r1[tid] = s; r2[tid] = q;
  __syncthreads();
  for (int off = 128; off > 0; off >>= 1) {
    if (tid < off) { r1[tid] += r1[tid + off]; r2[tid] += r2[tid + off]; }
    __syncthreads();
  }
  if (tid == 0) {
    float m = r1[0] * (1.0f / DDIM);
    float var = (r2[0] - (float)DDIM * m * m) * (1.0f / (DDIM - 1));
    var = fmaxf(var, 0.0f);
    float st = sqrtf(var);
    mu[b] = m; sd[b] = st;
    sMu = m; sInv = 1.0f / (st + LN_EPS);
  }
  __syncthreads();
  float4 o;
  o.x = (v.x - sMu) * sInv; o.y = (v.y - sMu) * sInv;
  o.z = (v.z - sMu) * sInv; o.w = (v.w - sMu) * sInv;
  *(float4*)(xn + (size_t)b * DDIM + tid * 4) = o;
}

// ------------------------------------------------------------------
// 2) Encoder row inverse L2 norms (folded into GEMM epilogue as col scale)
// ------------------------------------------------------------------
__global__ __launch_bounds__(256) void enc_norm_kernel(const float* __restrict__ wenc,
                                                       float* __restrict__ invn) {
  const int h = blockIdx.x;
  const int tid = threadIdx.x;
  const float4 v = *(const float4*)(wenc + (size_t)h * DDIM + tid * 4);
  float q = v.x * v.x + v.y * v.y + v.z * v.z + v.w * v.w;
  __shared__ float r[256];
  r[tid] = q;
  __syncthreads();
  for (int off = 128; off > 0; off >>= 1) {
    if (tid < off) r[tid] += r[tid + off];
    __syncthreads();
  }
  if (tid == 0) invn[h] = 1.0f / fmaxf(sqrtf(r[0]), 1e-12f);
}

// ------------------------------------------------------------------
// 3) Transpose W_dec [D][H] -> W_decT [H][D] so decode gathers are coalesced
// ------------------------------------------------------------------
__global__ __launch_bounds__(256) void transpose_kernel(const float* __restrict__ wdec,
                                                        float* __restrict__ wdt) {
  __shared__ float tile[32][33];
  const int tid = threadIdx.x;
  const int tx = tid & 31, ty = tid >> 5;
  const int h0 = blockIdx.x * 32, d0 = blockIdx.y * 32;
#pragma unroll
  for (int j = 0; j < 4; ++j) {
    int dr = ty + j * 8;
    tile[dr][tx] = wdec[(size_t)(d0 + dr) * HDIM + h0 + tx];
  }
  __syncthreads();
#pragma unroll
  for (int j = 0; j < 4; ++j) {
    int hr = ty + j * 8;
    wdt[(size_t)(h0 + hr) * DDIM + d0 + tx] = tile[tx][hr];
  }
}

// ------------------------------------------------------------------
// 4) Encoder GEMM: latents[b][h] = (xn[b,:] . W_enc[h,:]) * invn[h]
//    WMMA f32 16x16x4, block 128x128, 8 waves (2x4), wave tile 64x32.
//    Double-buffered LDS fed by GLOBAL_LOAD_ASYNC_TO_LDS_B128 (ASYNCcnt).
// ------------------------------------------------------------------
#define LDSTR 36  // 32 + 4 pad: 36*m mod 64 distinct for m in [0,16) -> conflict-free
#if USE_ASYNC_LDS
#define NBUF 2
#else
#define NBUF 1
#endif

__global__ __launch_bounds__(256) void enc_gemm_kernel(const float* __restrict__ xn,
                                                       const float* __restrict__ wenc,
                                                       const float* __restrict__ invn,
                                                       float* __restrict__ lat) {
  __shared__ float As[NBUF][128 * LDSTR];
  __shared__ float Bs[NBUF][128 * LDSTR];
  const int tid = threadIdx.x;
  const int gn0 = blockIdx.x * 128;  // H block
  const int gm0 = blockIdx.y * 128;  // B block
  const int lane = tid & 31, wave = tid >> 5;
  const int wm = wave >> 2;          // 0..1  -> 64-row slab
  const int wn = wave & 3;           // 0..3  -> 32-col slab
  const int lh = lane >> 4;          // half-wave selects K pair / M+8
  const int l15 = lane & 15;

  v8f acc[4][2] = {};

#if USE_ASYNC_LDS
  // Issue one 32-wide K chunk (8 async B128 per thread) into buffer `bf`.
  auto issue_chunk = [&](int k0, int bf) {
#pragma unroll
    for (int j = 0; j < 4; ++j) {
      int id = tid + j * 256;
      int row = id >> 3;
      int c4 = (id & 7) << 2;
      __builtin_amdgcn_global_load_async_to_lds_b128(
          (AS1 v4i*)(uintptr_t)(xn + (size_t)(gm0 + row) * DDIM + k0 + c4),
          (AS3 v4i*)(&As[bf][row * LDSTR + c4]), 0, 0);
      __builtin_amdgcn_global_load_async_to_lds_b128(
          (AS1 v4i*)(uintptr_t)(wenc + (size_t)(gn0 + row) * DDIM + k0 + c4),
          (AS3 v4i*)(&Bs[bf][row * LDSTR + c4]), 0, 0);
    }
  };
  int buf = 0;
  issue_chunk(0, 0);
#endif

  for (int k0 = 0; k0 < DDIM; k0 += 32) {
#if USE_ASYNC_LDS
    if (k0 + 32 < DDIM) {
      issue_chunk(k0 + 32, buf ^ 1);
      WAIT_ASYNCCNT(8);  // async loads retire in order: current buffer done
    } else {
      WAIT_ASYNCCNT(0);
    }
    __syncthreads();
    const float* __restrict__ AsP = As[buf];
    const float* __restrict__ BsP = Bs[buf];
#else
#pragma unroll
    for (int j = 0; j < 4; ++j) {
      int id = tid + j * 256;
      int row = id >> 3;
      int c4 = (id & 7) << 2;
      *(float4*)(As[0] + row * LDSTR + c4) =
          *(const float4*)(xn + (size_t)(gm0 + row) * DDIM + k0 + c4);
      *(float4*)(Bs[0] + row * LDSTR + c4) =
          *(const float4*)(wenc + (size_t)(gn0 + row) * DDIM + k0 + c4);
    }
    if (k0 + 32 < DDIM) {  // prefetch next K slab (global_prefetch_b8)
      int r = tid & 127;
      const float* p = (tid < 128)
          ? (wenc + (size_t)(gn0 + r) * DDIM + k0 + 32)
          : (xn + (size_t)(gm0 + r) * DDIM + k0 + 32);
      __builtin_prefetch(p, 0, 0);
    }
    __syncthreads();
    const float* __restrict__ AsP = As[0];
    const float* __restrict__ BsP = Bs[0];
#endif

#pragma unroll
    for (int k = 0; k < 32; k += 4) {
      const int kk = k + 2 * lh;  // lanes 0-15: K=k,k+1 ; lanes 16-31: K=k+2,k+3
      v2f bfr[2];
#pragma unroll
      for (int nt = 0; nt < 2; ++nt)
        bfr[nt] = *(const v2f*)(BsP + (wn * 32 + nt * 16 + l15) * LDSTR + kk);
#pragma unroll
      for (int mt = 0; mt < 4; ++mt) {
        v2f afr = *(const v2f*)(AsP + (wm * 64 + mt * 16 + l15) * LDSTR + kk);
#pragma unroll
        for (int nt = 0; nt < 2; ++nt)
          acc[mt][nt] = __builtin_amdgcn_wmma_f32_16x16x4_f32(
              false, afr, false, bfr[nt], (short)0, acc[mt][nt], false, false);
      }
    }
    __syncthreads();
#if USE_ASYNC_LDS
    buf ^= 1;
#endif
  }

  // Epilogue: scale by encoder inv-norm (column-wise), store latents.
#pragma unroll
  for (int nt = 0; nt < 2; ++nt) {
    const int n = gn0 + wn * 32 + nt * 16 + l15;
    const float rn = invn[n];
#pragma unroll
    for (int mt = 0; mt < 4; ++mt) {
      const int mb = gm0 + wm * 64 + mt * 16 + 8 * lh;
#pragma unroll
      for (int r = 0; r < 8; ++r)
        lat[(size_t)(mb + r) * HDIM + n] = acc[mt][nt][r] * rn;
    }
  }
}

// ------------------------------------------------------------------
// 5) Radix-select top-256 per row, compact (idx,val) pairs
// ------------------------------------------------------------------
__global__ __launch_bounds__(256) void topk_kernel(const float* __restrict__ lat,
                                                   unsigned* __restrict__ tdx,
                                                   float* __restrict__ tvl) {
  const int b = blockIdx.x;
  const int tid = threadIdx.x;
  const float* row = lat + (size_t)b * HDIM;
  __shared__ unsigned hist[256];
  __shared__ unsigned sPrefix, sK, sCnt, sEq;
  if (tid == 0) { sPrefix = 0u; sK = KSEL; }
  __syncthreads();
  for (int pass = 0; pass < 4; ++pass) {
    hist[tid] = 0u;
    __syncthreads();
    const int shift = 24 - 8 * pass;
    const unsigned pmask = pass ? (0xFFFFFFFFu << (shift + 8)) : 0u;
    const unsigned pref = sPrefix;
    for (int i = tid; i < HDIM; i += 256) {
      unsigned u = mapf(row[i]);
      if ((u & pmask) == pref) atomicAdd(&hist[(u >> shift) & 255u], 1u);
    }
    __syncthreads();
    if (tid == 0) {
      unsigned cum = 0, k = sK;
      for (int bin = 255; bin >= 0; --bin) {
        unsigned c = hist[bin];
        if (cum + c >= k) { sK = k - cum; sPrefix |= ((unsigned)bin) << shift; break; }
        cum += c;
      }
    }
    __syncthreads();
  }
  const unsigned kth = sPrefix;  // exact key of the 256th largest
  const unsigned rem = sK;       // how many ties at kth to keep
  if (tid == 0) { sCnt = 0u; sEq = 0u; }
  __syncthreads();
  for (int i = tid; i < HDIM; i += 256) {
    const float f = row[i];
    const unsigned u = mapf(f);
    if (u > kth) {
      unsigned pos = atomicAdd(&sCnt, 1u);
      tdx[(size_t)b * KSEL + pos] = (unsigned)i;
      tvl[(size_t)b * KSEL + pos] = f;
    } else if (u == kth) {
      unsigned t = atomicAdd(&sEq, 1u);
      if (t < rem) {
        unsigned pos = atomicAdd(&sCnt, 1u);
        tdx[(size_t)b * KSEL + pos] = (unsigned)i;
        tvl[(size_t)b * KSEL + pos] = f;
      }
    }
  }
}

// ------------------------------------------------------------------
// 6) Sparse decode: recons[b,:] = (sum_i val_i * W_decT[h_i,:]) * std[b] + mu[b]
// ------------------------------------------------------------------
__global__ __launch_bounds__(256) void decode_kernel(const unsigned* __restrict__ tdx,
                                                     const float* __restrict__ tvl,
                                                     const float* __restrict__ wdt,
                                                     const float* __restrict__ mu,
                                                     const float* __restrict__ sd,
                                                     float* __restrict__ recons) {
  const int b = blockIdx.x;
  const int tid = threadIdx.x;
  __shared__ unsigned sI[KSEL];
  __shared__ float sV[KSEL];
  sI[tid] = tdx[(size_t)b * KSEL + tid] & (HDIM - 1);
  sV[tid] = tvl[(size_t)b * KSEL + tid];
  __syncthreads();
  const int d0 = tid * 4;
  float a0 = 0.f, a1 = 0.f, a2 = 0.f, a3 = 0.f;
  for (int i = 0; i < KSEL; ++i) {
    const float v = sV[i];
    const float4 w = *(const float4*)(wdt + (size_t)sI[i] * DDIM + d0);
    a0 = fmaf(v, w.x, a0); a1 = fmaf(v, w.y, a1);
    a2 = fmaf(v, w.z, a2); a3 = fmaf(v, w.w, a3);
  }
  const float st = sd[b], m = mu[b];
  float4 o;
  o.x = fmaf(a0, st, m); o.y = fmaf(a1, st, m);
  o.z = fmaf(a2, st, m); o.w = fmaf(a3, st, m);
  *(float4*)(recons + (size_t)b * DDIM + d0) = o;
}

// ------------------------------------------------------------------
extern "C" void kernel_launch(void* const* d_in, const int* in_sizes, int n_in,
                              void* d_out, int out_size, void* d_ws, size_t ws_size,
                              hipStream_t stream) {
  const float* txt = (const float*)d_in[0];   // [2048,1024]
  const float* wenc = (const float*)d_in[1];  // [32768,1024]
  const float* wdec = (const float*)d_in[2];  // [1024,32768]
  float* out = (float*)d_out;
  float* recons = out;                          // [2048,1024]
  float* lat = out + (size_t)BDIM * DDIM;       // [2048,32768]

  // workspace carve-out (~147 MB)
  float* ws = (float*)d_ws;
  float* xn = ws;            ws += (size_t)BDIM * DDIM;   // 8 MB
  float* mu = ws;            ws += BDIM;
  float* sd = ws;            ws += BDIM;
  float* invn = ws;          ws += HDIM;                  // 128 KB
  float* wdt = ws;           ws += (size_t)HDIM * DDIM;   // 128 MB
  unsigned* tdx = (unsigned*)ws; ws += (size_t)BDIM * KSEL;  // 2 MB
  float* tvl = ws;                                         // 2 MB

  ln_kernel<<<BDIM, 256, 0, stream>>>(txt, xn, mu, sd);
  enc_norm_kernel<<<HDIM, 256, 0, stream>>>(wenc, invn);
  transpose_kernel<<<dim3(HDIM / 32, DDIM / 32), 256, 0, stream>>>(wdec, wdt);
  enc_gemm_kernel<<<dim3(HDIM / 128, BDIM / 128), 256, 0, stream>>>(xn, wenc, invn, lat);
  topk_kernel<<<BDIM, 256, 0, stream>>>(lat, tdx, tvl);
  decode_kernel<<<BDIM, 256, 0, stream>>>(tdx, tvl, wdt, mu, sd, recons);
}